// MessagePassingConvolution_2259152798305
// MI455X (gfx1250) — compile-verified
//
#include <hip/hip_runtime.h>
#include <hip/hip_bf16.h>
#include <math.h>

// CDNA5 (gfx1250) fused MACE-style message-passing convolution.
// Wave32: one wave = one 16-edge tile (matches WMMA M=16).
// MLP 8->64->64->64->96 done with V_WMMA_F32_16X16X4_F32 (exact fp32).
// silu / normalization use the HW transcendental unit (v_rcp_f32 / v_rsq_f32).

typedef float v2f __attribute__((ext_vector_type(2)));
typedef float v8f __attribute__((ext_vector_type(8)));

#define WAVES_PER_BLOCK 4
#define TILE_EDGES 16
#define H_STRIDE 97   // padded LDS row stride (floats): 97 mod 64 = 33 -> conflict-free column reads

__device__ __forceinline__ float silu_f(float x) {
    // sigmoid denominator is in (1, inf): hardware 1-ulp v_rcp_f32 is safe here
    return x * __builtin_amdgcn_rcpf(1.0f + __expf(-x));
}

// A fragment (16x4 f32): lane m = lane&15 holds row m; VGPR0 = K {0 | 2}, VGPR1 = K {1 | 3}
// for lane halves {0-15 | 16-31}.
__device__ __forceinline__ v2f load_a_lds(const float* __restrict__ h, int lane, int t) {
    const int m = lane & 15;
    const int k = t * 4 + ((lane >> 4) << 1);
    v2f a;
    a.x = h[m * H_STRIDE + k];
    a.y = h[m * H_STRIDE + k + 1];
    return a;
}

// B fragment (4x16 f32) from row-major W[IN][OUT]: lane holds column n*16 + (lane&15);
// VGPR0 = K {0 | 2}, VGPR1 = K {1 | 3} for lane halves.
__device__ __forceinline__ v2f load_b(const float* __restrict__ W, int lane, int t, int n, int OUT) {
    const int k = t * 4 + ((lane >> 4) << 1);
    const int col = n * 16 + (lane & 15);
    v2f b;
    b.x = W[k * OUT + col];
    b.y = W[(k + 1) * OUT + col];
    return b;
}

// C/D element r -> row (r + 8*(lane>=16)), col (lane&15) within the 16x16 tile.
__device__ __forceinline__ void store_c_lds(float* __restrict__ h, int lane, int n,
                                            v8f acc, float scale, bool act) {
    const int col = n * 16 + (lane & 15);
    const int rbase = (lane >> 4) << 3;
#pragma unroll
    for (int r = 0; r < 8; ++r) {
        float v = acc[r] * scale;
        if (act) v = silu_f(v);
        h[(rbase + r) * H_STRIDE + col] = v;
    }
}

__device__ __forceinline__ void gemm_layer(const float* __restrict__ hin,
                                           const float* __restrict__ W,
                                           float* __restrict__ hout,
                                           int IN, int OUT, float scale, bool act,
                                           int lane) {
    for (int n = 0; n < OUT / 16; ++n) {
        v8f acc = {};
        for (int t = 0; t < IN / 4; ++t) {
            v2f a = load_a_lds(hin, lane, t);
            v2f b = load_b(W, lane, t, n, OUT);
            acc = __builtin_amdgcn_wmma_f32_16x16x4_f32(false, a, false, b,
                                                        (short)0, acc, false, false);
        }
        store_c_lds(hout, lane, n, acc, scale, act);
    }
}

__global__ __launch_bounds__(128, 1)
void mpconv_fused_kernel(const float* __restrict__ vectors,
                         const float* __restrict__ node_feats,
                         const float* __restrict__ radial,
                         const int* __restrict__ senders,
                         const int* __restrict__ receivers,
                         const float* __restrict__ w0,
                         const float* __restrict__ w1,
                         const float* __restrict__ w2,
                         const float* __restrict__ w3,
                         float* __restrict__ out,
                         int E) {
    __shared__ float lds_hA[WAVES_PER_BLOCK][TILE_EDGES * H_STRIDE];
    __shared__ float lds_hB[WAVES_PER_BLOCK][TILE_EDGES * H_STRIDE];
    __shared__ float lds_s[WAVES_PER_BLOCK][TILE_EDGES][32];
    __shared__ float lds_y[WAVES_PER_BLOCK][TILE_EDGES][8];

    const int lane = (int)(threadIdx.x & 31);
    const int wave = (int)(threadIdx.x >> 5);
    const int tile = (int)blockIdx.x * WAVES_PER_BLOCK + wave;
    const int base = tile * TILE_EDGES;

    float* hA = lds_hA[wave];
    float* hB = lds_hB[wave];

    // ---- Layer 0: radial(16x8) @ w0(8x64), scale 1/sqrt(8), silu -> hA ----
    {
        const float scale = 0.35355339059327373f;
        const int m = lane & 15;
        int e = base + m;
        if (e >= E) e = E - 1;
        const int khalf = (lane >> 4) << 1;
        for (int n = 0; n < 4; ++n) {
            v8f acc = {};
#pragma unroll
            for (int t = 0; t < 2; ++t) {
                v2f a;
                a.x = radial[e * 8 + t * 4 + khalf];
                a.y = radial[e * 8 + t * 4 + khalf + 1];
                v2f b = load_b(w0, lane, t, n, 64);
                acc = __builtin_amdgcn_wmma_f32_16x16x4_f32(false, a, false, b,
                                                            (short)0, acc, false, false);
            }
            store_c_lds(hA, lane, n, acc, scale, true);
        }
    }
    __syncthreads();
    // ---- Layer 1: hA(16x64) @ w1(64x64), scale 1/8, silu -> hB ----
    gemm_layer(hA, w1, hB, 64, 64, 0.125f, true, lane);
    __syncthreads();
    // ---- Layer 2: hB @ w2 -> hA ----
    gemm_layer(hB, w2, hA, 64, 64, 0.125f, true, lane);
    __syncthreads();
    // ---- Layer 3: hA @ w3(64x96), scale 1/8, no act -> hB (the "mix") ----
    gemm_layer(hA, w3, hB, 64, 96, 0.125f, false, lane);
    __syncthreads();

    // ---- Stage sender features (coalesced 128B rows) ----
    for (int e2 = 0; e2 < TILE_EDGES; ++e2) {
        int e = base + e2;
        if (e >= E) e = E - 1;
        const int snd = senders[e];
        lds_s[wave][e2][lane] = node_feats[(size_t)snd * 32 + lane];
    }

    // ---- Spherical harmonics per edge (lanes 0..15) ----
    if (lane < TILE_EDGES) {
        int e = base + lane;
        if (e >= E) e = E - 1;
        const float vx = -vectors[e * 3 + 0];
        const float vy = -vectors[e * 3 + 1];
        const float vz = -vectors[e * 3 + 2];
        const float rn = __builtin_amdgcn_rsqf(vx * vx + vy * vy + vz * vz + 1e-20f);
        const float x = vx * rn, y = vy * rn, z = vz * rn;
        const float s3  = 1.7320508075688772f;
        const float s15 = 3.872983346207417f;
        const float s5  = 2.23606797749979f;
        float* yp = lds_y[wave][lane];
        yp[0] = s3 * x;
        yp[1] = s3 * y;
        yp[2] = s3 * z;
        yp[3] = s15 * x * z;
        yp[4] = s15 * x * y;
        yp[5] = s5 * (y * y - 0.5f * (x * x + z * z));
        yp[6] = s15 * y * z;
        yp[7] = 0.5f * s15 * (z * z - x * x);
    }
    __syncthreads();

    // ---- Scatter messages: 288 columns per edge, 9 columns per lane ----
    const float inv_avg = 0.125f; // 1 / AVG_NUM_NEIGHBORS
    for (int e2 = 0; e2 < TILE_EDGES; ++e2) {
        const int e = base + e2;
        if (e >= E) break; // uniform across wave (tile either whole or tail)
        const int rcv = receivers[e];
        float* orow = out + (size_t)rcv * 288;
        const float* srow = lds_s[wave][e2];
        const float* yrow = lds_y[wave][e2];
        const float* mixr = hB + e2 * H_STRIDE;
#pragma unroll
        for (int jj = 0; jj < 9; ++jj) {
            const int j = lane + jj * 32;
            float val;
            if (j < 32) {
                val = srow[j] * mixr[j];
            } else if (j < 128) {
                const int q = j - 32;
                const int c = q / 3, k = q - c * 3;
                val = srow[c] * yrow[k] * mixr[32 + c];
            } else {
                const int q = j - 128;
                const int c = q / 5, k = q - c * 5;
                val = srow[c] * yrow[3 + k] * mixr[64 + c];
            }
            atomicAdd(orow + j, val * inv_avg);
        }
    }
}

extern "C" void kernel_launch(void* const* d_in, const int* in_sizes, int n_in,
                              void* d_out, int out_size, void* d_ws, size_t ws_size,
                              hipStream_t stream) {
    const float* vectors    = (const float*)d_in[0];
    const float* node_feats = (const float*)d_in[1];
    const float* radial     = (const float*)d_in[2];
    const int*   senders    = (const int*)d_in[3];
    const int*   receivers  = (const int*)d_in[4];
    const float* w0         = (const float*)d_in[5];
    const float* w1         = (const float*)d_in[6];
    const float* w2         = (const float*)d_in[7];
    const float* w3         = (const float*)d_in[8];
    float* out = (float*)d_out;

    const int E = in_sizes[0] / 3;

    // Output is accumulated with atomics -> must start from zero every call.
    hipMemsetAsync(d_out, 0, (size_t)out_size * sizeof(float), stream);

    const int tiles  = (E + TILE_EDGES - 1) / TILE_EDGES;
    const int blocks = (tiles + WAVES_PER_BLOCK - 1) / WAVES_PER_BLOCK;
    mpconv_fused_kernel<<<blocks, WAVES_PER_BLOCK * 32, 0, stream>>>(
        vectors, node_feats, radial, senders, receivers, w0, w1, w2, w3, out, E);
}